// YOLOv1Loss_80384607912704
// MI455X (gfx1250) — compile-verified
//
#include <hip/hip_runtime.h>
#include <hip/hip_bf16.h>

// ---------------------------------------------------------------------------
// YOLOv1 loss on MI455X (gfx1250).
// Roofline: 193 MB of f32 input @ 23.3 TB/s => ~8.3 us; ~80 MFLOP total.
// Strictly bandwidth bound -> coalesced staging into LDS (async-to-LDS b128
// on CDNA5: no VGPR round trip, tracked by ASYNCcnt), one cell per thread,
// two-pass deterministic reduction. Wave reduction implemented with
// V_WMMA_F32_16X16X4_F32 (A = partials, B = ones) per CDNA5 ISA 7.12.2.
// ---------------------------------------------------------------------------

typedef __attribute__((ext_vector_type(2))) float v2f;
typedef __attribute__((ext_vector_type(8))) float v8f;
typedef int v4i __attribute__((vector_size(16)));   // matches builtin param type

#define TILE      256
#define CELL_F    30
#define TILE_F    (TILE * CELL_F)   // 7680 floats per tensor per tile
#define TILE_F4   (TILE_F / 4)      // 1920 float4s

#define AS1 __attribute__((address_space(1)))
#define AS3 __attribute__((address_space(3)))

#if __has_builtin(__builtin_amdgcn_global_load_async_to_lds_b128)
#define HAS_ASYNC_LDS 1
#else
#define HAS_ASYNC_LDS 0
#endif

// Full 32-lane wave sum via one WMMA.
// A-matrix 16x4 f32 layout (ISA 7.12.2): lane m (m<16) holds A[m][0],A[m][1];
// lane m+16 holds A[m][2],A[m][3].  With B == ones(4x16), C == 0:
//   D[m][n] = rowsum[m] = acc_m + acc_{m+16}           (A.y carries 0)
// D layout: lane L<16, vgpr v -> D[v][L]; lane L>=16 -> D[v+8][L-16].
// So sum_{v=0..7} d[v] gives (lanes 0-15)  sum_{m=0..7}  rowsum[m]
//                        and (lanes 16-31) sum_{m=8..15} rowsum[m];
// one shfl_xor(16) + add completes the 32-lane reduction in every lane.
__device__ __forceinline__ float wave_reduce_wmma(float acc) {
    v2f a; a.x = acc;  a.y = 0.0f;
    v2f b; b.x = 1.0f; b.y = 1.0f;
    v8f c = {};
    v8f d = __builtin_amdgcn_wmma_f32_16x16x4_f32(
        /*neg_a=*/false, a, /*neg_b=*/false, b,
        /*c_mod=*/(short)0, c, /*reuse_a=*/false, /*reuse_b=*/false);
    float h = d[0] + d[1] + d[2] + d[3] + d[4] + d[5] + d[6] + d[7];
    h += __shfl_xor(h, 16);
    return h;
}

__global__ void __launch_bounds__(TILE)
yolo_pass1(const float* __restrict__ pred,
           const float* __restrict__ tgt,
           float* __restrict__ partials,
           int nCells) {
    __shared__ __align__(16) float sP[TILE_F];
    __shared__ __align__(16) float sT[TILE_F];
    __shared__ float red[TILE / 32];

    const int tid       = threadIdx.x;
    const int cellBase  = blockIdx.x * TILE;
    const int tileCells = min(TILE, nCells - cellBase);
    const int nFl       = tileCells * CELL_F;

    const float* gP = pred + (size_t)cellBase * CELL_F;  // tile base: 30720B multiple -> 16B aligned
    const float* gT = tgt  + (size_t)cellBase * CELL_F;

    if (nFl == TILE_F) {
        // Fully coalesced b128 staging (hot path; tiling is exact for 16384*49 cells).
        const float4* gP4 = reinterpret_cast<const float4*>(gP);
        const float4* gT4 = reinterpret_cast<const float4*>(gT);
        float4* sP4 = reinterpret_cast<float4*>(sP);
        float4* sT4 = reinterpret_cast<float4*>(sT);
#if HAS_ASYNC_LDS
        // CDNA5 async global->LDS DMA: no VGPR landing, tracked by ASYNCcnt.
        for (int i = tid; i < TILE_F4; i += TILE) {
            __builtin_amdgcn_global_load_async_to_lds_b128(
                (AS1 v4i*)(gP4 + i), (AS3 v4i*)(sP4 + i), 0, 0);
            __builtin_amdgcn_global_load_async_to_lds_b128(
                (AS1 v4i*)(gT4 + i), (AS3 v4i*)(sT4 + i), 0, 0);
        }
#if __has_builtin(__builtin_amdgcn_s_wait_asynccnt)
        __builtin_amdgcn_s_wait_asynccnt(0);
#else
        asm volatile("s_wait_asynccnt 0" ::: "memory");
#endif
#else
        for (int i = tid; i < TILE_F4; i += TILE) {
            sP4[i] = gP4[i];
            sT4[i] = gT4[i];
        }
#endif
    } else {
        for (int i = tid; i < nFl; i += TILE) { sP[i] = gP[i]; sT[i] = gT[i]; }
    }
    __syncthreads();

    float acc = 0.0f;
    if (tid < tileCells) {
        const float* p = sP + tid * CELL_F;   // stride-30 LDS reads: bank-conflict free in wave32
        const float* t = sT + tid * CELL_F;

        const float t4  = t[4];
        const float coo = (t4 > 0.0f) ? 1.0f : 0.0f;
        const float noo = 1.0f - coo;

        // no-object confidence loss (both boxes' confidences)
        const float d4 = p[4] - t4;
        const float d9 = p[9] - t[9];
        const float nooobj = noo * (d4 * d4 + d9 * d9);

        // class loss over channels 10..29
        float cls = 0.0f;
        #pragma unroll
        for (int ch = 10; ch < 30; ++ch) {
            const float dd = p[ch] - t[ch];
            cls += dd * dd;
        }
        cls *= coo;

        // target box 0 corners (mirror reference's exact arithmetic)
        const float tx = t[0], ty = t[1], tw = t[2], th = t[3];
        const float tltx = tx - 0.5f * tw, tlty = ty - 0.5f * th;
        const float trbx = tx + 0.5f * tw, trby = ty + 0.5f * th;
        const float area2 = (trbx - tltx) * (trby - tlty);

        float iou0 = 0.0f, iou1 = 0.0f;
        #pragma unroll
        for (int bx = 0; bx < 2; ++bx) {
            const float px = p[bx * 5 + 0], py = p[bx * 5 + 1];
            const float pw = p[bx * 5 + 2], ph = p[bx * 5 + 3];
            const float pltx = px - 0.5f * pw, plty = py - 0.5f * ph;
            const float prbx = px + 0.5f * pw, prby = py + 0.5f * ph;
            const float ltx = fmaxf(pltx, tltx), lty = fmaxf(plty, tlty);
            const float rbx = fminf(prbx, trbx), rby = fminf(prby, trby);
            // reference quirk: 0/1 indicator of (rb - lt < 0), NOT overlap area
            const float wi = (rbx - ltx < 0.0f) ? 1.0f : 0.0f;
            const float hi = (rby - lty < 0.0f) ? 1.0f : 0.0f;
            const float inter = wi * hi;
            const float area1 = (prbx - pltx) * (prby - plty);
            const float iou = inter / (area1 + area2 - inter);
            if (bx == 0) iou0 = iou; else iou1 = iou;
        }
        // jnp.argmax: first index on ties
        const int idx = (iou1 > iou0) ? 1 : 0;
        const float* rp = p + idx * 5;
        const float* rt = t + idx * 5;

        const float dc = rp[4] - rt[4];
        const float contain = coo * dc * dc;

        const bool safe = coo > 0.0f;
        const float swpx = sqrtf(safe ? rp[2] : 1.0f);
        const float swpy = sqrtf(safe ? rp[3] : 1.0f);
        const float swtx = sqrtf(safe ? rt[2] : 1.0f);
        const float swty = sqrtf(safe ? rt[3] : 1.0f);
        const float dx  = rp[0] - rt[0], dy  = rp[1] - rt[1];
        const float dsx = swpx - swtx,   dsy = swpy - swty;
        const float loc = coo * (dx * dx + dy * dy + dsx * dsx + dsy * dsy);

        acc = 5.0f * loc + contain + 0.5f * nooobj + cls;
    }
    // EXEC is all-ones here (the guarded region above is closed) -> WMMA legal.
    const float wsum = wave_reduce_wmma(acc);

    const int lane = tid & 31;
    const int wid  = tid >> 5;
    if (lane == 0) red[wid] = wsum;
    __syncthreads();
    if (tid == 0) {
        float bs = 0.0f;
        #pragma unroll
        for (int w = 0; w < TILE / 32; ++w) bs += red[w];
        partials[blockIdx.x] = bs;
    }
}

__global__ void yolo_pass2(const float* __restrict__ partials, int n,
                           float* __restrict__ out, float invBatch) {
    float s = 0.0f;
    for (int i = (int)threadIdx.x; i < n; i += 32) s += partials[i];
    const float tot = wave_reduce_wmma(s);
    if (threadIdx.x == 0) out[0] = tot * invBatch;
}

extern "C" void kernel_launch(void* const* d_in, const int* in_sizes, int n_in,
                              void* d_out, int out_size, void* d_ws, size_t ws_size,
                              hipStream_t stream) {
    const float* pred = (const float*)d_in[0];
    const float* tgt  = (const float*)d_in[1];

    const int nFloats = in_sizes[0];          // N*S*S*30
    const int nCells  = nFloats / CELL_F;     // N*S*S = 802816
    const int nBatch  = nCells / 49;          // N = 16384
    const int nBlocks = (nCells + TILE - 1) / TILE;   // 3136 (exact tiling)

    float* partials = (float*)d_ws;           // nBlocks * 4 B = 12.5 KB scratch

    yolo_pass1<<<nBlocks, TILE, 0, stream>>>(pred, tgt, partials, nCells);
    yolo_pass2<<<1, 32, 0, stream>>>(partials, nBlocks, (float*)d_out,
                                     1.0f / (float)nBatch);
}